// GCN_26568667693833
// MI455X (gfx1250) — compile-verified
//
#include <hip/hip_runtime.h>
#include <hip/hip_bf16.h>

typedef float v2f __attribute__((ext_vector_type(2)));
typedef float v8f __attribute__((ext_vector_type(8)));

#define IN_CH 128
#define HID   64
#define WLDS_STRIDE 160  // 128 data floats + 32 pad => lane halves hit disjoint bank halves

// ---------------- degree / normalization ----------------
__global__ void k_init_deg(float* __restrict__ deg, int n) {
  int i = blockIdx.x * blockDim.x + threadIdx.x;
  if (i < n) deg[i] = 1.0f;  // self-loop fill_value = 1
}

__global__ void k_deg_scatter(const int* __restrict__ col, const float* __restrict__ ew,
                              float* __restrict__ deg, int e) {
  int i = blockIdx.x * blockDim.x + threadIdx.x;
  if (i < e) atomicAdd(&deg[col[i]], ew[i]);
}

__global__ void k_dinv(float* __restrict__ deg, int n) {
  int i = blockIdx.x * blockDim.x + threadIdx.x;
  if (i < n) {
    float d = deg[i];
    deg[i] = (d > 0.0f) ? rsqrtf(d) : 0.0f;
  }
}

__global__ void k_zero(float* __restrict__ p, size_t n) {
  size_t i = (size_t)blockIdx.x * blockDim.x + threadIdx.x;
  size_t stride = (size_t)gridDim.x * blockDim.x;
  for (; i < n; i += stride) p[i] = 0.0f;
}

// ---------------- layer-1 GEMM: h = x @ W1 (fp32 WMMA) ----------------
// One wave per 16-row tile; 4 x (16x16) output tiles cover HID=64 columns.
// W1 staged in LDS pre-paired for the WMMA B layout so each B fragment is a
// single contiguous ds_load_b64 (no VGPR packing moves):
//   Wlds[kp * WLDS_STRIDE + n*2 + j] = W1[(2*kp + j)*HID + n]
__global__ __launch_bounds__(256)
void k_gemm_wmma(const float* __restrict__ x, const float* __restrict__ W1g,
                 float* __restrict__ h, int n) {
  __shared__ float Wlds[(IN_CH / 2) * WLDS_STRIDE];  // 40 KB
  for (int idx = threadIdx.x; idx < IN_CH * HID; idx += 256) {
    int k  = idx >> 6;       // row of W1 (coalesced global read)
    int nn = idx & (HID - 1);
    Wlds[(k >> 1) * WLDS_STRIDE + nn * 2 + (k & 1)] = W1g[idx];
  }
  __syncthreads();

  const int wave = threadIdx.x >> 5;
  const int lane = threadIdx.x & 31;
  const int tileRow = (blockIdx.x * 8 + wave) * 16;
  if (tileRow >= n) return;

  const int m  = lane & 15;          // A: row within tile; B/D: column within tile
  const int kb = (lane >> 4) << 1;   // A/B: K-pair selected by lane half

  int row = tileRow + m;
  if (row > n - 1) row = n - 1;      // clamp loads; stores are guarded
  const float* xrow  = x + (size_t)row * IN_CH;
  const float* wbase = &Wlds[(lane >> 4) * WLDS_STRIDE + m * 2];

  v8f acc[4] = {v8f{}, v8f{}, v8f{}, v8f{}};
  for (int k = 0; k < IN_CH; k += 4) {
    // A fragment: lane holds A[m][k+kb], A[m][k+kb+1]  (contiguous 8B load)
    v2f a = *(const v2f*)(xrow + k + kb);
    const float* wk = wbase + (k >> 1) * WLDS_STRIDE;
#pragma unroll
    for (int t = 0; t < 4; ++t) {
      // B fragment: contiguous pair {W1[k+kb][t*16+m], W1[k+kb+1][t*16+m]}
      v2f b = *(const v2f*)(wk + t * 32);
      acc[t] = __builtin_amdgcn_wmma_f32_16x16x4_f32(
          /*neg_a=*/false, a, /*neg_b=*/false, b,
          /*c_mod=*/(short)0, acc[t], /*reuse_a=*/false, /*reuse_b=*/false);
    }
  }

  // D layout: lane -> N = m, VGPR v -> M = v + 8*(lane>=16)
  const int mh = (lane >> 4) << 3;
  float* hbase = h + (size_t)(tileRow + mh) * HID + m;
  if (tileRow + 16 <= n) {
    // Full tile (the common case; N % 16 == 0 makes this every live tile):
    // unguarded stores off one base address with immediate offsets.
#pragma unroll
    for (int t = 0; t < 4; ++t) {
#pragma unroll
      for (int v = 0; v < 8; ++v) {
        hbase[(size_t)v * HID + t * 16] = acc[t][v];
      }
    }
  } else {
    // Partial last tile: per-row guard.
#pragma unroll
    for (int t = 0; t < 4; ++t) {
#pragma unroll
      for (int v = 0; v < 8; ++v) {
        if (tileRow + mh + v < n) hbase[(size_t)v * HID + t * 16] = acc[t][v];
      }
    }
  }
}

// ---------------- layer-1 edge aggregation ----------------
// One wave32 per edge; each lane handles features (lane, lane+32).
__global__ __launch_bounds__(256)
void k_scatter1(const int* __restrict__ row, const int* __restrict__ col,
                const float* __restrict__ ew, const float* __restrict__ dinv,
                const float* __restrict__ h, float* __restrict__ agg, int e) {
  const int wave = threadIdx.x >> 5;
  const int lane = threadIdx.x & 31;
  const long long ei = (long long)blockIdx.x * 8 + wave;
  if (ei >= e) return;
  const int r = row[ei];
  const int c = col[ei];
  const float norm = dinv[r] * ew[ei] * dinv[c];
  const float m0 = h[(size_t)r * HID + lane] * norm;
  const float m1 = h[(size_t)r * HID + lane + 32] * norm;
  atomicAdd(&agg[(size_t)c * HID + lane], m0);
  atomicAdd(&agg[(size_t)c * HID + lane + 32], m1);
}

// ---------------- finalize layer-1 + start layer-2 ----------------
// h1 = relu(agg + h*dinv^2 + b1); s = h1 . W2; out = s*dinv^2 + b2 (self term)
__global__ __launch_bounds__(256)
void k_finalize1(const float* __restrict__ agg, const float* __restrict__ hpre,
                 const float* __restrict__ dinv, const float* __restrict__ b1,
                 const float* __restrict__ W2, const float* __restrict__ b2,
                 float* __restrict__ s, float* __restrict__ out, int n) {
  const int wave = threadIdx.x >> 5;
  const int lane = threadIdx.x & 31;
  const int i = blockIdx.x * 8 + wave;
  if (i >= n) return;
  const float di = dinv[i];
  const float d2 = di * di;
  const size_t base = (size_t)i * HID;
  float v0 = agg[base + lane]      + hpre[base + lane]      * d2 + b1[lane];
  float v1 = agg[base + lane + 32] + hpre[base + lane + 32] * d2 + b1[lane + 32];
  v0 = v0 > 0.0f ? v0 : 0.0f;
  v1 = v1 > 0.0f ? v1 : 0.0f;
  float p = v0 * W2[lane] + v1 * W2[lane + 32];
#pragma unroll
  for (int off = 16; off > 0; off >>= 1) p += __shfl_xor(p, off, 32);
  if (lane == 0) {
    s[i] = p;
    out[i] = p * d2 + b2[0];
  }
}

// ---------------- layer-2 edge aggregation (scalar feature) ----------------
__global__ void k_scatter2(const int* __restrict__ row, const int* __restrict__ col,
                           const float* __restrict__ ew, const float* __restrict__ dinv,
                           const float* __restrict__ s, float* __restrict__ out, int e) {
  int i = blockIdx.x * blockDim.x + threadIdx.x;
  if (i < e) {
    const int r = row[i];
    const int c = col[i];
    atomicAdd(&out[c], s[r] * dinv[r] * ew[i] * dinv[c]);
  }
}

extern "C" void kernel_launch(void* const* d_in, const int* in_sizes, int n_in,
                              void* d_out, int out_size, void* d_ws, size_t ws_size,
                              hipStream_t stream) {
  (void)n_in; (void)out_size; (void)ws_size;
  const float* x  = (const float*)d_in[0];
  const int*   ei = (const int*)  d_in[1];
  const float* ew = (const float*)d_in[2];
  const float* W1 = (const float*)d_in[3];
  const float* b1 = (const float*)d_in[4];
  const float* W2 = (const float*)d_in[5];
  const float* b2 = (const float*)d_in[6];

  const int N = in_sizes[0] / IN_CH;
  const int E = in_sizes[2];
  const int* row = ei;        // edge_index[0]
  const int* col = ei + E;    // edge_index[1]
  float* out = (float*)d_out;

  // workspace carve (256B aligned): dinv[N], h[N*64], agg[N*64], s[N]
  char* ws = (char*)d_ws;
  const size_t nA  = (((size_t)N * 4)       + 255) & ~(size_t)255;
  const size_t nhA = (((size_t)N * HID * 4) + 255) & ~(size_t)255;
  float* dinv = (float*)ws;
  float* hbuf = (float*)(ws + nA);
  float* agg  = (float*)(ws + nA + nhA);
  float* sbuf = (float*)(ws + nA + 2 * nhA);

  k_init_deg   <<<(N + 255) / 256, 256, 0, stream>>>(dinv, N);
  k_deg_scatter<<<(E + 255) / 256, 256, 0, stream>>>(col, ew, dinv, E);
  k_dinv       <<<(N + 255) / 256, 256, 0, stream>>>(dinv, N);
  k_zero       <<<2048, 256, 0, stream>>>(agg, (size_t)N * HID);
  k_gemm_wmma  <<<(N + 127) / 128, 256, 0, stream>>>(x, W1, hbuf, N);
  k_scatter1   <<<(E + 7) / 8, 256, 0, stream>>>(row, col, ew, dinv, hbuf, agg, E);
  k_finalize1  <<<(N + 7) / 8, 256, 0, stream>>>(agg, hbuf, dinv, b1, W2, b2, sbuf, out, N);
  k_scatter2   <<<(E + 255) / 256, 256, 0, stream>>>(row, col, ew, dinv, sbuf, out, E);
}